// FreqGrid_52235392254249
// MI455X (gfx1250) — compile-verified
//
#include <hip/hip_runtime.h>

typedef __attribute__((ext_vector_type(2))) float v2f;
typedef __attribute__((ext_vector_type(8))) float v8f;

#define RR 512
#define NC 8
#define NFQ 9           // base freqs
#define NI 17           // 2*NFQ-1 output freq slots
#define THREADS 256
#define PTS_PER_BLK 32  // THREADS / NC

__global__ __launch_bounds__(THREADS) void freqgrid_kernel(
    const float* __restrict__ coords, const float* __restrict__ grid,
    const float* __restrict__ Wexp,   const float* __restrict__ bexp,
    const float* __restrict__ freqs,  float* __restrict__ out)
{
    __shared__ float s_basis[PTS_PER_BLK * NI * 3];   // 1632 f
    __shared__ float s_coef[8][32][NFQ * 3];          // per-wave staging (27/lane)
    __shared__ float s_exp [8][32][(NFQ - 1) * 3];    // per-wave results (24/lane)
    __shared__ float s_trash[16 * 8];                 // sink for zero rows g=8..15

    const int tid  = threadIdx.x;
    const int lane = tid & 31;
    const int wv   = tid >> 5;
    const int p    = blockIdx.x * THREADS + tid;      // (n,c) pair index
    const int n    = p >> 3;
    const int c    = p & 7;
    const int n0   = blockIdx.x * PTS_PER_BLK;

    // ---------- Phase 0: per-point cosine basis into LDS (shared over c) ----------
    for (int idx = tid; idx < PTS_PER_BLK * NI * 3; idx += THREADS) {
        int pl  = idx / (NI * 3);
        int rem = idx - pl * (NI * 3);
        int i   = rem / 3;
        int d   = rem - i * 3;
        float crd   = coords[(size_t)(n0 + pl) * 3 + d];
        float phase = (3.14159265358979323846f / (float)RR) *
                      ((crd + 1.0f) * 0.5f * (float)(RR - 1) + 0.5f);
        float fr   = fminf(fmaxf(freqs[i], 0.0f), 1.0f);
        float feff = exp2f(fr * 9.0f) - 1.0f;         // log2(512) = 9
        s_basis[idx] = cosf(phase * (feff + 0.5f));
    }
    __syncthreads();

    // ---------- Phase 1: bilinear gather of 27 coefs (the real cost) ----------
    const float cx = coords[(size_t)n * 3 + 0];
    const float cy = coords[(size_t)n * 3 + 1];
    const float cz = coords[(size_t)n * 3 + 2];
    const float sx[3] = {cy, cx, cx};                 // plane x = pts[:,0]
    const float sy[3] = {cz, cz, cy};                 // plane y = pts[:,1]

    float coef[NFQ][3];
    #pragma unroll
    for (int k = 0; k < 3; ++k) {
        float fx  = (sx[k] + 1.0f) * 0.5f * (float)(RR - 1);
        float fy  = (sy[k] + 1.0f) * 0.5f * (float)(RR - 1);
        float fx0 = fminf(fmaxf(floorf(fx), 0.0f), (float)(RR - 1));
        float fy0 = fminf(fmaxf(floorf(fy), 0.0f), (float)(RR - 1));
        int x0 = (int)fx0, y0 = (int)fy0;
        int x1 = (x0 + 1 < RR) ? x0 + 1 : RR - 1;
        int y1 = (y0 + 1 < RR) ? y0 + 1 : RR - 1;
        float wx = fx - fx0, wy = fy - fy0;
        float w00 = (1.0f - wy) * (1.0f - wx), w01 = (1.0f - wy) * wx;
        float w10 = wy * (1.0f - wx),          w11 = wy * wx;
        int i00 = y0 * RR + x0, i01 = y0 * RR + x1;
        int i10 = y1 * RR + x0, i11 = y1 * RR + x1;
        int chbase = (k * (NC * NFQ) + c * NFQ) * (RR * RR);
        #pragma unroll
        for (int f = 0; f < NFQ; ++f) {
            int b = chbase + f * (RR * RR);
            float v = grid[b + i00] * w00 + grid[b + i01] * w01 +
                      grid[b + i10] * w10 + grid[b + i11] * w11;
            coef[f][k] = v;
            s_coef[wv][lane][f * 3 + k] = v;
        }
    }
    __syncthreads();

    // ---------- Phase 2: frequency expansion via V_WMMA_F32_16X16X4_F32 ----------
    // D(16x16) = A(16x4 slice of W_exp, rows g) x B(4x16, cols = 16 (lane,k) triples)
    // No exec-mask manipulation anywhere here: all loads unconditional (clamped
    // in-bounds indices), tail-chunk zeros produced with value-selects only.
    const int half = lane >> 4;   // selects K pair {0,1} vs {2,3}
    const int mrow = lane & 15;
    const int grow = (mrow < 8) ? mrow : 7;           // clamped W_exp row (in-bounds)
    const float rowmask = (mrow < 8) ? 1.0f : 0.0f;   // zero rows g=8..15
    v2f A0, A1, A2;
    {
        int f0 = half * 2;        // chunk0: f = 0..3
        int f4 = 4 + half * 2;    // chunk1: f = 4..7
        A0[0] = Wexp[grow * NFQ + f0]     * rowmask;
        A0[1] = Wexp[grow * NFQ + f0 + 1] * rowmask;
        A1[0] = Wexp[grow * NFQ + f4]     * rowmask;
        A1[1] = Wexp[grow * NFQ + f4 + 1] * rowmask;
        // chunk2: only f=8 (half==0, first reg) is real; rest zero-pad
        float a8 = Wexp[grow * NFQ + 8] * rowmask;
        A2[0] = (half == 0) ? a8 : 0.0f;
        A2[1] = 0.0f;
    }
    #pragma unroll
    for (int t = 0; t < 6; ++t) {                 // 6 tiles: (k=t>>1, half-wave h=t&1)
        const int k = t >> 1, h = t & 1;
        const int src = h * 16 + mrow;            // owner lane of this column
        int f0 = half * 2, f4 = 4 + half * 2;
        v2f B0, B1, B2;
        B0[0] = s_coef[wv][src][f0 * 3 + k];
        B0[1] = s_coef[wv][src][(f0 + 1) * 3 + k];
        B1[0] = s_coef[wv][src][f4 * 3 + k];
        B1[1] = s_coef[wv][src][(f4 + 1) * 3 + k];
        float b8 = s_coef[wv][src][8 * 3 + k];    // in-bounds for all lanes
        B2[0] = (half == 0) ? b8 : 0.0f;          // value-select, not exec mask
        B2[1] = 0.0f;
        v8f d = {0.0f, 0.0f, 0.0f, 0.0f, 0.0f, 0.0f, 0.0f, 0.0f};
        d = __builtin_amdgcn_wmma_f32_16x16x4_f32(false, A0, false, B0, (short)0, d, false, false);
        d = __builtin_amdgcn_wmma_f32_16x16x4_f32(false, A1, false, B1, (short)0, d, false, false);
        d = __builtin_amdgcn_wmma_f32_16x16x4_f32(false, A2, false, B2, (short)0, d, false, false);
        // lanes 0-15 hold rows g=0..7 (valid); lanes 16-31 hold zero rows -> trash.
        // Ternary ADDRESS (not a branch) keeps EXEC all-ones around WMMA.
        float* dst = (lane < 16) ? &s_exp[wv][src][k * 8] : &s_trash[mrow * 8];
        #pragma unroll
        for (int g = 0; g < 8; ++g) dst[g] = d[g];
    }
    __syncthreads();

    // ---------- Phase 3: interleave, modulate by basis, write 51 floats ----------
    const int pl = tid >> 3;                      // local point (= n - n0)
    float* op = out + (size_t)n * (NC * NI * 3) + (size_t)c * (NI * 3);
    const float* bs = &s_basis[pl * (NI * 3)];
    #pragma unroll
    for (int j = 0; j < NFQ; ++j)
        #pragma unroll
        for (int k = 0; k < 3; ++k)
            op[(2 * j) * 3 + k] = 2.0f * coef[j][k] * bs[(2 * j) * 3 + k];
    #pragma unroll
    for (int g = 0; g < NFQ - 1; ++g) {
        float bg = bexp[g];
        #pragma unroll
        for (int k = 0; k < 3; ++k) {
            float e = s_exp[wv][lane][k * 8 + g] + bg;
            op[(2 * g + 1) * 3 + k] = 2.0f * e * bs[(2 * g + 1) * 3 + k];
        }
    }
}

extern "C" void kernel_launch(void* const* d_in, const int* in_sizes, int n_in,
                              void* d_out, int out_size, void* d_ws, size_t ws_size,
                              hipStream_t stream) {
    const float* coords = (const float*)d_in[0];
    const float* grid   = (const float*)d_in[1];
    const float* Wexp   = (const float*)d_in[2];
    const float* bexp   = (const float*)d_in[3];
    const float* freqs  = (const float*)d_in[4];
    float* out = (float*)d_out;
    const int N = in_sizes[0] / 3;                // 524288
    const int blocks = (N * NC) / THREADS;        // N*8 divisible by 256
    hipLaunchKernelGGL(freqgrid_kernel, dim3(blocks), dim3(THREADS), 0, stream,
                       coords, grid, Wexp, bexp, freqs, out);
}